// LIFSpike_65395172049718
// MI455X (gfx1250) — compile-verified
//
#include <hip/hip_runtime.h>
#include <hip/hip_bf16.h>

// LIF spiking recurrence, memory-bound streaming kernel for MI455X (gfx1250).
// x:   [B=32, T=10, C=128, H=32, W=32] fp32
// out: spikes, same shape, fp32 (0.0 / 1.0)
//
// Data path: global -> LDS via async DMA (GLOBAL_LOAD_ASYNC_TO_LDS_B128,
// tracked by ASYNCcnt), LDS -> VGPR via ds_load_b128, spikes -> global via
// non-temporal 128-bit stores. 335.5 MB total HBM traffic => ~14.4 us floor
// at 23.3 TB/s.

typedef float v4f __attribute__((ext_vector_type(4)));

#define T_STEPS 10
#define S4      32768      // C*H*W/4 = 128*32*32/4 float4 columns per (b,t) slice
#define TAU     0.5f
#define THRESH  1.0f
#define BLOCK   256

__global__ __launch_bounds__(BLOCK) void lif_spike_kernel(
    const float* __restrict__ x, float* __restrict__ out, int total4) {
  // One float4 slot per (thread, timestep): 256 * 10 * 16B = 40 KB.
  __shared__ v4f smem[T_STEPS * BLOCK];

  const int tid4 = blockIdx.x * BLOCK + threadIdx.x;
  if (tid4 >= total4) return;  // grid is exact; guard never taken

  const int b  = tid4 >> 15;          // tid4 / S4
  const int i4 = tid4 & (S4 - 1);     // tid4 % S4
  const size_t base = (size_t)b * (size_t)(T_STEPS * S4) + (size_t)i4;

  const v4f* __restrict__ in4  = (const v4f*)x;
  v4f* __restrict__       out4 = (v4f*)out;

  // ---- Issue all T async global->LDS 128-bit DMA copies (fire & forget). ----
  // LDS address is taken from the vdst VGPR; low 32 bits of the flat shared pointer
  // are the LDS byte offset. Loads complete in order (ASYNCcnt pipeline).
#pragma unroll
  for (int t = 0; t < T_STEPS; ++t) {
    unsigned lds_addr = (unsigned)(size_t)(&smem[t * BLOCK + threadIdx.x]);
    const v4f* gptr = in4 + base + (size_t)t * (size_t)S4;
    asm volatile("global_load_async_to_lds_b128 %0, %1, off"
                 :
                 : "v"(lds_addr), "v"(gptr)
                 : "memory");
  }

  // ---- LIF recurrence; overlap compute on step t with DMA of t+1..T-1. ----
  v4f mem = {0.0f, 0.0f, 0.0f, 0.0f};
#pragma unroll
  for (int t = 0; t < T_STEPS; ++t) {
    // Async loads retire in order: ASYNCcnt <= (T-1-t) means slice t landed.
    asm volatile("s_wait_asynccnt %0" : : "i"(T_STEPS - 1 - t) : "memory");

    const v4f xt = smem[t * BLOCK + threadIdx.x];   // ds_load_b128

    v4f spk;
#pragma unroll
    for (int c = 0; c < 4; ++c) {
      float m = __builtin_fmaf(mem[c], TAU, xt[c]); // mem = tau*mem + x_t
      bool fired = (m > THRESH);                    // mem - 1.0 > 0
      spk[c] = fired ? 1.0f : 0.0f;
      mem[c] = fired ? 0.0f : m;                    // hard reset
    }

    // Streamed output, never re-read: bypass cache retention (TH_STORE_NT).
    __builtin_nontemporal_store(spk, &out4[base + (size_t)t * (size_t)S4]);
  }
}

extern "C" void kernel_launch(void* const* d_in, const int* in_sizes, int n_in,
                              void* d_out, int out_size, void* d_ws, size_t ws_size,
                              hipStream_t stream) {
  const float* x = (const float*)d_in[0];
  float* out = (float*)d_out;

  // in_sizes[0] = B*T*C*H*W = 41,943,040; one thread per float4 column
  // (time handled inside the thread): total4 = B*C*H*W/4.
  const int total4 = in_sizes[0] / (4 * T_STEPS);
  const int grid = (total4 + BLOCK - 1) / BLOCK;

  lif_spike_kernel<<<grid, BLOCK, 0, stream>>>(x, out, total4);
}